// PINNLoss_35304631173376
// MI455X (gfx1250) — compile-verified
//
#include <hip/hip_runtime.h>
#include <hip/hip_bf16.h>
#include <math.h>

// ---------------------------------------------------------------------------
// PINN loss for 6-DoF rocket dynamics, MI455X (gfx1250, wave32).
// Single streaming pass (HBM-bound, ~277MB -> ~12us floor @ 23.3 TB/s),
// per-wave reduction done with V_WMMA_F32_16X16X4_F32 (B = ones trick),
// block partials -> ws, tiny finalize kernel combines 6 outputs.
// ---------------------------------------------------------------------------

typedef __attribute__((ext_vector_type(2))) float v2f;
typedef __attribute__((ext_vector_type(8))) float v8f;

#define BSZ     256
#define NSZ     8192
#define SDIM    14
#define NBLOCKS 1024
#define TPB     256
#define NACC    6

// Full wave32 sum of one f32 value per lane, via WMMA with all-ones B.
// D[m][n] = sum_k A[m][k]  (B==1 everywhere). Each lane then sums its 8
// D VGPRs (= sum over half the M rows) and adds the xor-16 partner's sum.
// Result: total sum replicated to every lane; independent of exact A layout.
__device__ __forceinline__ float wave_sum_wmma(float acc) {
  v2f a; a.x = acc;  a.y = 0.0f;
  v2f b; b.x = 1.0f; b.y = 1.0f;
  v8f c = {};
  v8f d = __builtin_amdgcn_wmma_f32_16x16x4_f32(
      /*neg_a=*/false, a, /*neg_b=*/false, b,
      /*c_mod=*/(short)0, c, /*reuse_a=*/false, /*reuse_b=*/false);
  float s = d[0] + d[1] + d[2] + d[3] + d[4] + d[5] + d[6] + d[7];
  s += __shfl_xor(s, 16, 32);
  return s;
}

__global__ __launch_bounds__(TPB) void pinn_main(
    const float* __restrict__ pred, const float* __restrict__ truth,
    const float* __restrict__ t,    const float* __restrict__ ctrl,
    float* __restrict__ partials) {
  const float EPS = 1e-12f;
  const float g0 = 9.80665f, Isp = 300.0f, rho = 1.225f, Cd = 0.5f, Ar = 1.0f;
  const float T_max = 1.0e6f, tau_max = 1.0e4f;
  const float Ixx = 5000.0f, Iyy = 5000.0f, Izz = 1000.0f;

  // accumulators: [trans, rot, mass, phys, quat, massflow]
  float accT = 0.f, accR = 0.f, accM = 0.f, accP = 0.f, accQ = 0.f, accF = 0.f;

  const long long total = (long long)BSZ * NSZ;
  for (long long i = (long long)blockIdx.x * TPB + threadIdx.x; i < total;
       i += (long long)NBLOCKS * TPB) {
    const int b = (int)(i / NSZ);
    const int n = (int)(i % NSZ);

    const float* sp = pred  + i * SDIM;
    const float* st = truth + i * SDIM;
    float s[SDIM], sv[SDIM];
#pragma unroll
    for (int k = 0; k < 7; ++k) {  // rows are 8B-aligned (56B stride)
      float2 v = ((const float2*)sp)[k]; s [2*k] = v.x; s [2*k+1] = v.y;
      float2 w = ((const float2*)st)[k]; sv[2*k] = w.x; sv[2*k+1] = w.y;
    }

    // ---- data loss (COMP_W == 1) ----
    float dtr = 0.f, dro = 0.f;
#pragma unroll
    for (int k = 0; k < 6; ++k)  { float d = s[k] - sv[k]; dtr += d * d; }
#pragma unroll
    for (int k = 6; k < 13; ++k) { float d = s[k] - sv[k]; dro += d * d; }
    { float d = s[13] - sv[13]; accM += d * d; }
    accT += dtr; accR += dro;

    // ---- quaternion norm loss ----
    const float q0 = s[6], q1 = s[7], q2 = s[8], q3 = s[9];
    float qn = sqrtf(q0*q0 + q1*q1 + q2*q2 + q3*q3);
    accQ += (qn - 1.f) * (qn - 1.f);

    // ---- neighbor rows for finite difference ----
    const int nm = (n > 0)       ? n - 1 : n;
    const int np = (n < NSZ - 1) ? n + 1 : n;
    const float* spm = pred + ((long long)b * NSZ + nm) * SDIM;
    const float* spp = pred + ((long long)b * NSZ + np) * SDIM;
    float sm[SDIM], sq[SDIM];
#pragma unroll
    for (int k = 0; k < 7; ++k) {
      float2 v = ((const float2*)spm)[k]; sm[2*k] = v.x; sm[2*k+1] = v.y;
      float2 w = ((const float2*)spp)[k]; sq[2*k] = w.x; sq[2*k+1] = w.y;
    }

    // ---- mass-flow loss (relu(mass[n+1] - mass[n]), n < N-1) ----
    if (n < NSZ - 1) accF += fmaxf(sq[13] - s[13], 0.f);

    // ---- finite-difference dt (t[0,:,0]) ----
    const float tn = t[n], tmv = t[nm], tpv = t[np];
    const bool first = (n == 0), last = (n == NSZ - 1);
    const float dt = first ? (tpv - tn) : (last ? (tn - tmv) : (tpv - tmv));
    const float rdt = 1.f / (dt + EPS);

    // ---- rigid-body dynamics ----
    const float4 u4 = *((const float4*)(ctrl + i * 4));  // 16B aligned
    const float thrust = u4.x * T_max;
    const float taux = u4.y * tau_max, tauy = u4.z * tau_max, tauz = u4.w * tau_max;
    const float vx = s[3], vy = s[4], vz = s[5];
    const float wx = s[10], wy = s[11], wz = s[12];
    const float m = s[13];

    const float bzx = 2.f * (q1 * q3 + q0 * q2);
    const float bzy = 2.f * (q2 * q3 - q0 * q1);
    const float bzz = 1.f - 2.f * (q1 * q1 + q2 * q2);
    const float speed = sqrtf(vx*vx + vy*vy + vz*vz + EPS);
    const float dragc = -0.5f * rho * Cd * Ar * speed;
    const float invm = 1.f / m;

    float dyn[SDIM];
    dyn[0] = vx; dyn[1] = vy; dyn[2] = vz;
    dyn[3] = (thrust * bzx + dragc * vx) * invm;
    dyn[4] = (thrust * bzy + dragc * vy) * invm;
    dyn[5] = (thrust * bzz + dragc * vz) * invm - g0;
    dyn[6] = 0.5f * (-q1 * wx - q2 * wy - q3 * wz);
    dyn[7] = 0.5f * ( q0 * wx + q2 * wz - q3 * wy);
    dyn[8] = 0.5f * ( q0 * wy - q1 * wz + q3 * wx);
    dyn[9] = 0.5f * ( q0 * wz + q1 * wy - q2 * wx);
    dyn[10] = (taux - (Izz - Iyy) * wy * wz) / Ixx;
    dyn[11] = (tauy - (Ixx - Izz) * wz * wx) / Iyy;
    dyn[12] = (tauz - (Iyy - Ixx) * wx * wy) / Izz;
    dyn[13] = -thrust / (Isp * g0);

    float ph = 0.f;
#pragma unroll
    for (int k = 0; k < SDIM; ++k) {
      const float lov = first ? s[k] : sm[k];
      const float hiv = last  ? s[k] : sq[k];
      const float r = (hiv - lov) * rdt - dyn[k];
      ph += r * r;
    }
    accP += ph;
  }

  // ---- per-wave reduction via WMMA, then per-block via LDS ----
  __shared__ float smem[TPB / 32][NACC];
  const int wave = threadIdx.x >> 5;
  const int lane = threadIdx.x & 31;
  const float r0 = wave_sum_wmma(accT);
  const float r1 = wave_sum_wmma(accR);
  const float r2 = wave_sum_wmma(accM);
  const float r3 = wave_sum_wmma(accP);
  const float r4 = wave_sum_wmma(accQ);
  const float r5 = wave_sum_wmma(accF);
  if (lane == 0) {
    smem[wave][0] = r0; smem[wave][1] = r1; smem[wave][2] = r2;
    smem[wave][3] = r3; smem[wave][4] = r4; smem[wave][5] = r5;
  }
  __syncthreads();
  if (threadIdx.x == 0) {
    float o[NACC] = {0.f, 0.f, 0.f, 0.f, 0.f, 0.f};
#pragma unroll
    for (int w = 0; w < TPB / 32; ++w)
#pragma unroll
      for (int j = 0; j < NACC; ++j) o[j] += smem[w][j];
    float* outp = partials + (long long)blockIdx.x * NACC;
#pragma unroll
    for (int j = 0; j < NACC; ++j) outp[j] = o[j];
  }
}

__global__ __launch_bounds__(256) void pinn_finalize(
    const float* __restrict__ partials, const float* __restrict__ t,
    const float* __restrict__ pred,     const float* __restrict__ truth,
    float* __restrict__ out) {
  __shared__ double sred[256];
  __shared__ float  svals[256];
  __shared__ int    sidx[256];
  __shared__ int    bcidx;
  const int tid = threadIdx.x;

  // stage 1: sum the 1024 block partials (double accumulation)
  double comp[NACC];
  for (int j = 0; j < NACC; ++j) {
    double a = 0.0;
    for (int k = tid; k < NBLOCKS; k += 256) a += (double)partials[k * NACC + j];
    sred[tid] = a;
    __syncthreads();
    for (int off = 128; off > 0; off >>= 1) {
      if (tid < off) sred[tid] += sred[tid + off];
      __syncthreads();
    }
    comp[j] = sred[0];
    __syncthreads();
  }

  // stage 2: argmin_n |t[0,n,0]| (first occurrence of the minimum)
  float mv = 3.4e38f; int mi = 0;
  for (int n = tid; n < NSZ; n += 256) {
    float v = fabsf(t[n]);
    if (v < mv) { mv = v; mi = n; }
  }
  svals[tid] = mv; sidx[tid] = mi;
  __syncthreads();
  for (int off = 128; off > 0; off >>= 1) {
    if (tid < off) {
      if (svals[tid + off] < svals[tid] ||
          (svals[tid + off] == svals[tid] && sidx[tid + off] < sidx[tid])) {
        svals[tid] = svals[tid + off]; sidx[tid] = sidx[tid + off];
      }
    }
    __syncthreads();
  }
  if (tid == 0) bcidx = sidx[0];
  __syncthreads();
  const int idx = bcidx;

  // stage 3: boundary-condition loss at time index idx
  double bca = 0.0;
  for (int j = tid; j < BSZ * SDIM; j += 256) {
    const int b = j / SDIM, c = j % SDIM;
    const float d = pred[((long long)b * NSZ + idx) * SDIM + c] -
                    truth[((long long)b * NSZ + idx) * SDIM + c];
    bca += (double)d * (double)d;
  }
  sred[tid] = bca;
  __syncthreads();
  for (int off = 128; off > 0; off >>= 1) {
    if (tid < off) sred[tid] += sred[tid + off];
    __syncthreads();
  }

  if (tid == 0) {
    const double BN = (double)BSZ * (double)NSZ;
    const double L_data = comp[0] / (BN * 6.0) + comp[1] / (BN * 7.0) + comp[2] / BN;
    const double L_phys = comp[3] / (BN * 14.0);
    const double L_quat = comp[4] / BN;
    const double L_mf   = comp[5] / ((double)BSZ * (double)(NSZ - 1));
    const double L_bc   = sred[0] / ((double)BSZ * (double)SDIM);
    const double total  = 1.0 * L_data + 0.1 * L_phys + 1.0 * L_bc +
                          0.01 * L_quat + 0.01 * L_mf;
    out[0] = (float)total; out[1] = (float)L_data; out[2] = (float)L_phys;
    out[3] = (float)L_bc;  out[4] = (float)L_quat; out[5] = (float)L_mf;
  }
}

extern "C" void kernel_launch(void* const* d_in, const int* in_sizes, int n_in,
                              void* d_out, int out_size, void* d_ws, size_t ws_size,
                              hipStream_t stream) {
  const float* pred  = (const float*)d_in[0];   // (256, 8192, 14) f32
  const float* truth = (const float*)d_in[1];   // (256, 8192, 14) f32
  const float* t     = (const float*)d_in[2];   // (256, 8192, 1)  f32
  const float* ctrl  = (const float*)d_in[3];   // (256, 8192, 4)  f32
  float* partials = (float*)d_ws;               // NBLOCKS * NACC floats

  pinn_main<<<NBLOCKS, TPB, 0, stream>>>(pred, truth, t, ctrl, partials);
  pinn_finalize<<<1, 256, 0, stream>>>(partials, t, pred, truth, (float*)d_out);
}